// RegularNonlinearity_20615843021563
// MI455X (gfx1250) — compile-verified
//
#include <hip/hip_runtime.h>

// RegularNonlinearity: per (batch,channel) 6-pt irfft -> relu -> rfft.
// out7 = F(7x6) * relu( G(6x8) * v8 ), G/F constant; mapped onto
// V_WMMA_F32_16X16X4_F32 with 16 channels per wave-tile (N dim), K=8 split
// into two K=4 WMMA steps per stage. Memory-bound: ~0.94 GB @ 23.3 TB/s.

typedef float v2f __attribute__((ext_vector_type(2)));
typedef float v8f __attribute__((ext_vector_type(8)));

#define NUMCH 256
#define FEAT  1792   // 256 + 2*256*3

#define S3  0.8660254037844386f   // sqrt(3)/2
#define S36 0.2886751345948129f   // sqrt(3)/6

// Inverse-DFT matrix G (6x8, rows 6..15 zero). Columns (K): re0, im0(=0),
// re1, im1, re2, im2, re3, pad.  signal[t] = (1/6)(re0 + 2*Re(z1 w^t)
// + 2*Re(z2 w^2t) + re3*(-1)^t),  w = e^{i*pi/3}.
__device__ __constant__ float G8[16][8] = {
    {1.f/6.f, 0.f,  1.f/3.f,  0.f,   1.f/3.f,  0.f,   1.f/6.f, 0.f},
    {1.f/6.f, 0.f,  1.f/6.f, -S36,  -1.f/6.f, -S36,  -1.f/6.f, 0.f},
    {1.f/6.f, 0.f, -1.f/6.f, -S36,  -1.f/6.f,  S36,   1.f/6.f, 0.f},
    {1.f/6.f, 0.f, -1.f/3.f,  0.f,   1.f/3.f,  0.f,  -1.f/6.f, 0.f},
    {1.f/6.f, 0.f, -1.f/6.f,  S36,  -1.f/6.f, -S36,   1.f/6.f, 0.f},
    {1.f/6.f, 0.f,  1.f/6.f,  S36,  -1.f/6.f,  S36,  -1.f/6.f, 0.f},
};

// Forward-rfft matrix F (7x6 padded to 16x8; cols 6,7 zero). Rows (M):
// re0', re1', im1', re2', im2', re3', im3'(=0).
__device__ __constant__ float F8[16][8] = {
    {1.f,  1.f,  1.f,  1.f,  1.f,  1.f, 0.f, 0.f},
    {1.f,  0.5f, -0.5f, -1.f, -0.5f, 0.5f, 0.f, 0.f},
    {0.f, -S3,  -S3,   0.f,  S3,   S3,  0.f, 0.f},
    {1.f, -0.5f, -0.5f, 1.f, -0.5f, -0.5f, 0.f, 0.f},
    {0.f, -S3,   S3,   0.f, -S3,   S3,  0.f, 0.f},
    {1.f, -1.f,  1.f, -1.f,  1.f, -1.f, 0.f, 0.f},
};

__global__ __launch_bounds__(256) void
RegularNonlinearity_kernel(const float* __restrict__ x,
                           float* __restrict__ y, int nrows) {
    const int row  = blockIdx.x;
    if (row >= nrows) return;
    const int lane = threadIdx.x & 31;
    const int wave = threadIdx.x >> 5;
    const int m    = lane & 15;            // M row / task-in-tile index
    const bool low = lane < 16;
    const int kb   = low ? 0 : 2;          // A-layout: VGPR0=K0|K2, VGPR1=K1|K3

    // Hoisted A operands (constant matrices in WMMA A layout, 2 VGPRs each)
    v2f gA0 = {G8[m][kb],     G8[m][kb + 1]};     // G cols 0..3
    v2f gA1 = {G8[m][4 + kb], G8[m][5 + kb]};     // G cols 4..7
    v2f fA0 = {F8[m][kb],     F8[m][kb + 1]};     // F cols 0..3
    v2f fA1 = {F8[m][4 + kb], F8[m][5 + kb]};     // F cols 4..7

    const float* xr = x + (size_t)row * FEAT;
    float*       yr = y + (size_t)row * FEAT;

#pragma unroll
    for (int iter = 0; iter < 2; ++iter) {
        const int c = (wave + iter * 8) * 16 + m;   // channel for this lane

        // Streaming non-temporal loads (0.9 GB stream >> 192 MB L2).
        // Both half-waves load the same channel tile -> coalesced duplicates.
        float r0 = __builtin_nontemporal_load(xr + c);
        v2f z1 = __builtin_nontemporal_load(
            reinterpret_cast<const v2f*>(xr + 256) + c);   // (re1,im1)
        v2f z2 = __builtin_nontemporal_load(
            reinterpret_cast<const v2f*>(xr + 768) + c);   // (re2,im2)
        v2f z3 = __builtin_nontemporal_load(
            reinterpret_cast<const v2f*>(xr + 1280) + c);  // (re3,im3)

        // B operand (8x16): VGPR0 = K0|K2, VGPR1 = K1|K3 across half-waves.
        v2f b01 = { low ? r0   : z1.x,    // re0 | re1
                    low ? 0.f  : z1.y };  // im0 | im1
        v2f b23 = { low ? z2.x : z3.x,    // re2 | re3
                    low ? z2.y : 0.f  };  // im2 | pad (im3 dropped by irfft)

        // Stage 1: signal = G * v   (two K=4 f32 WMMA steps)
        v8f s = {};
        s = __builtin_amdgcn_wmma_f32_16x16x4_f32(false, gA0, false, b01,
                                                  (short)0, s, false, false);
        s = __builtin_amdgcn_wmma_f32_16x16x4_f32(false, gA1, false, b23,
                                                  (short)0, s, false, false);

        // ReLU on accumulator (rows 6..15 are exact zeros already)
#pragma unroll
        for (int i = 0; i < 8; ++i) s[i] = fmaxf(s[i], 0.f);

        // Re-pack D (row t in VGPR t, lanes 0-15) into B layout for stage 2:
        // rows 2,3 must move to lanes 16-31.
        float s2 = __shfl_xor(s[2], 16, 32);
        float s3 = __shfl_xor(s[3], 16, 32);
        v2f c01 = { low ? s[0] : s2, low ? s[1] : s3 };   // K0|K2, K1|K3
        v2f c23 = { low ? s[4] : 0.f, low ? s[5] : 0.f }; // K4|K6(=0), K5|K7(=0)

        // Stage 2: out = F * relu(signal)
        v8f o = {};
        o = __builtin_amdgcn_wmma_f32_16x16x4_f32(false, fA0, false, c01,
                                                  (short)0, o, false, false);
        o = __builtin_amdgcn_wmma_f32_16x16x4_f32(false, fA1, false, c23,
                                                  (short)0, o, false, false);

        // D layout: lane l (l<16) holds all 7 outputs of its channel in o[0..6].
        if (low) {
            __builtin_nontemporal_store(o[0], yr + c);
            v2f w1 = {o[1], o[2]};   // re1', im1'
            v2f w2 = {o[3], o[4]};   // re2', im2'
            v2f w3 = {o[5], o[6]};   // re3', im3'(=0)
            __builtin_nontemporal_store(w1, reinterpret_cast<v2f*>(yr + 256) + c);
            __builtin_nontemporal_store(w2, reinterpret_cast<v2f*>(yr + 768) + c);
            __builtin_nontemporal_store(w3, reinterpret_cast<v2f*>(yr + 1280) + c);
        }
    }
}

extern "C" void kernel_launch(void* const* d_in, const int* in_sizes, int n_in,
                              void* d_out, int out_size, void* d_ws, size_t ws_size,
                              hipStream_t stream) {
    (void)n_in; (void)out_size; (void)d_ws; (void)ws_size;
    const float* x = (const float*)d_in[0];
    float* y = (float*)d_out;
    const int nrows = in_sizes[0] / FEAT;   // 65536
    RegularNonlinearity_kernel<<<nrows, 256, 0, stream>>>(x, y, nrows);
}